// Aggregate_24223615550063
// MI455X (gfx1250) — compile-verified
//
#include <hip/hip_runtime.h>
#include <hip/hip_bf16.h>
#include <math.h>

// ---------------------------------------------------------------------------
// Problem constants (match reference)
// ---------------------------------------------------------------------------
#define NUM_CAT      16
#define NUM_ATTR     144
#define NUM_NUMERIC  128      // NUM_ATTR - NUM_CAT
#define N_ARY        32
#define NUM_SEGMENTS 100000

typedef float v2f __attribute__((ext_vector_type(2)));
typedef float v8f __attribute__((ext_vector_type(8)));

// Workspace layout (d_ws):
//   [0   ..131]  int cfg[33]   : cfg[0..31]=top_num_idx, cfg[32]=top_cat_idx
//   [256 ..383]  float conf[32]
//   [512 .. )    float agg[NUM_SEGMENTS*32]  (compacted segment-sum, 12.8MB)

// ---------------------------------------------------------------------------
// Phase 0: softmax + top-k of the two masks (tiny; one thread is plenty).
// ---------------------------------------------------------------------------
__global__ void prep_kernel(const float* __restrict__ cat_mask,
                            const float* __restrict__ num_mask,
                            int* __restrict__ cfg,
                            float* __restrict__ conf) {
  if (blockIdx.x != 0 || threadIdx.x != 0) return;

  // category softmax -> top-1 (value = 1/sum(exp(x-max)) at the argmax)
  float cmax = cat_mask[0]; int cidx = 0;
  for (int i = 1; i < NUM_CAT; ++i) {
    float v = cat_mask[i];
    if (v > cmax) { cmax = v; cidx = i; }
  }
  float csum = 0.f;
  for (int i = 0; i < NUM_CAT; ++i) csum += expf(cat_mask[i] - cmax);
  float top_cat_val = 1.0f / csum;
  cfg[32] = cidx;

  // numeric softmax
  float nmax = num_mask[0];
  for (int i = 1; i < NUM_NUMERIC; ++i) nmax = fmaxf(nmax, num_mask[i]);
  float p[NUM_NUMERIC];
  float nsum = 0.f;
  for (int i = 0; i < NUM_NUMERIC; ++i) { p[i] = expf(num_mask[i] - nmax); nsum += p[i]; }
  float inv = 1.0f / nsum;

  // top-32 by selection (strict '>' matches top_k tie-break: lower index first)
  bool used[NUM_NUMERIC];
  for (int i = 0; i < NUM_NUMERIC; ++i) used[i] = false;
  for (int j = 0; j < N_ARY; ++j) {
    int best = 0; float bv = -INFINITY;
    for (int i = 0; i < NUM_NUMERIC; ++i)
      if (!used[i] && p[i] > bv) { bv = p[i]; best = i; }
    used[best] = true;
    cfg[j]  = best;
    conf[j] = 0.5f * (bv * inv + top_cat_val);
  }
}

// ---------------------------------------------------------------------------
// Phase 1: zero the compacted aggregation table (3.2M floats). Regular-
// temporal stores on purpose: we WANT these lines resident in L2 for the
// atomic storm that follows.
// ---------------------------------------------------------------------------
__global__ void zero_kernel(float4* __restrict__ agg4, int n4) {
  int i = blockIdx.x * blockDim.x + threadIdx.x;
  int stride = gridDim.x * blockDim.x;
  float4 z = make_float4(0.f, 0.f, 0.f, 0.f);
  for (; i < n4; i += stride) agg4[i] = z;
}

// ---------------------------------------------------------------------------
// Phase 2: compacted atomic scatter. One wave per row-iteration:
// lane j owns selected column j. 32M L2-resident f32 atomics total.
// inputs (512MB, read-once) loaded non-temporal so they don't evict the
// agg table / idx lines from the 192MB L2; seg id load kept scalar+RT.
// ---------------------------------------------------------------------------
__global__ void scatter_kernel(const float* __restrict__ inputs,
                               const int*   __restrict__ idx,
                               const int*   __restrict__ cfg,
                               float*       __restrict__ agg,
                               int nrows) {
  const int lane   = threadIdx.x & 31;
  const int wave   = (blockIdx.x * blockDim.x + threadIdx.x) >> 5;
  const int nwaves = (gridDim.x * blockDim.x) >> 5;

  const int sel    = cfg[lane];                               // per-lane column
  const int catcol = __builtin_amdgcn_readfirstlane(cfg[32]); // wave-uniform SGPR

  for (int r = wave; r < nrows; r += nwaves) {
    // keep the HBM stream ahead: prefetch next iteration's row region
    if (r + nwaves < nrows)
      __builtin_prefetch(inputs + (size_t)(r + nwaves) * NUM_ATTR + NUM_CAT + sel, 0, 1);
    const int seg   = idx[r * NUM_CAT + catcol];  // uniform -> s_load, stays in L2
    const float val =
        __builtin_nontemporal_load(inputs + (size_t)r * NUM_ATTR + NUM_CAT + sel);
    atomicAdd(&agg[(size_t)seg * N_ARY + lane], val);  // global_atomic_add_f32
  }
}

// ---------------------------------------------------------------------------
// Phase 3: gather + conf-scale as WMMA: out(16x32) = A(16x32) * diag(conf).
// V_WMMA_F32_16X16X4_F32, exact f32. One wave = 16 rows, EXEC all-ones.
// A layout (ISA 7.12.2, 16x4 f32): lane L: M=L%16; v0->K=0|2, v1->K=1|3
// B layout (4x16 f32):             lane L: N=L%16; v0->K=0|2, v1->K=1|3
// D layout (16x16 f32, 8 VGPRs):   vgpr i: M=i+8*(L>=16), N=L%16
// Output is write-once -> non-temporal stores.
// ---------------------------------------------------------------------------
__global__ void gather_wmma_kernel(const float* __restrict__ agg,
                                   const int*   __restrict__ idx,
                                   const int*   __restrict__ cfg,
                                   const float* __restrict__ conf,
                                   float*       __restrict__ out,
                                   int ntiles) {
  const int lane = threadIdx.x & 31;
  const int wave = (blockIdx.x * blockDim.x + threadIdx.x) >> 5;
  if (wave >= ntiles) return;            // uniform per wave; EXEC stays all-ones

  const int n  = lane & 15;              // column within 16-wide group / row for A
  const int hi = lane >> 4;              // 0: K=0..1 half, 1: K=2..3 half
  const int rowBase = wave * 16;
  const int catcol  = __builtin_amdgcn_readfirstlane(cfg[32]);

  const int seg = idx[(rowBase + n) * NUM_CAT + catcol];
  const float* __restrict__ arow = agg + (size_t)seg * N_ARY;

#pragma unroll
  for (int g = 0; g < 2; ++g) {          // column groups [0,16) and [16,32)
    const float cf = conf[16 * g + n];
    v8f acc = {};
#pragma unroll
    for (int c = 0; c < 4; ++c) {        // K = 16 per group, 4 chunks of 4
      const int kb = 4 * c + 2 * hi;     // this lane-half's K pair within group
      v2f a = *(const v2f*)(arow + 16 * g + kb);   // A[M=n, K=kb..kb+1]
      v2f b;
      b.x = (n == kb)     ? cf : 0.f;    // diag(conf) slice in B layout
      b.y = (n == kb + 1) ? cf : 0.f;
      acc = __builtin_amdgcn_wmma_f32_16x16x4_f32(
          /*neg_a=*/false, a, /*neg_b=*/false, b,
          /*c_mod=*/(short)0, acc, /*reuse_a=*/false, /*reuse_b=*/false);
    }
#pragma unroll
    for (int i = 0; i < 8; ++i)
      __builtin_nontemporal_store(
          acc[i], out + (size_t)(rowBase + i + 8 * hi) * N_ARY + 16 * g + n);
  }
}

// Scalar tail for row counts not divisible by 16 (unused for N_ROWS=1e6).
__global__ void gather_tail_kernel(const float* __restrict__ agg,
                                   const int*   __restrict__ idx,
                                   const int*   __restrict__ cfg,
                                   const float* __restrict__ conf,
                                   float*       __restrict__ out,
                                   int row0, int nrows) {
  const int lane = threadIdx.x & 31;
  const int wave = (blockIdx.x * blockDim.x + threadIdx.x) >> 5;
  const int r = row0 + wave;
  if (r >= nrows) return;
  const int seg = idx[r * NUM_CAT + __builtin_amdgcn_readfirstlane(cfg[32])];
  out[(size_t)r * N_ARY + lane] = agg[(size_t)seg * N_ARY + lane] * conf[lane];
}

// ---------------------------------------------------------------------------
// Host launch
// ---------------------------------------------------------------------------
extern "C" void kernel_launch(void* const* d_in, const int* in_sizes, int n_in,
                              void* d_out, int out_size, void* d_ws, size_t ws_size,
                              hipStream_t stream) {
  const float* inputs   = (const float*)d_in[0];
  const int*   idx      = (const int*)  d_in[1];
  const float* cat_mask = (const float*)d_in[2];
  const float* num_mask = (const float*)d_in[3];
  float*       out      = (float*)d_out;

  const int nrows = in_sizes[0] / NUM_ATTR;

  int*   cfg  = (int*)  d_ws;
  float* conf = (float*)((char*)d_ws + 256);
  float* agg  = (float*)((char*)d_ws + 512);

  prep_kernel<<<1, 32, 0, stream>>>(cat_mask, num_mask, cfg, conf);

  const int n4 = (NUM_SEGMENTS * N_ARY) / 4;   // 800k float4
  zero_kernel<<<1024, 256, 0, stream>>>((float4*)agg, n4);

  scatter_kernel<<<2048, 256, 0, stream>>>(inputs, idx, cfg, agg, nrows);

  const int ntiles = nrows >> 4;               // 62500 waves, 16 rows each
  if (ntiles > 0) {
    const int waves_per_block = 4;             // blockDim 128 = 4 wave32
    gather_wmma_kernel<<<(ntiles + waves_per_block - 1) / waves_per_block,
                         waves_per_block * 32, 0, stream>>>(
        agg, idx, cfg, conf, out, ntiles);
  }
  const int rem = nrows - (ntiles << 4);
  if (rem > 0) {
    gather_tail_kernel<<<(rem + 7) / 8, 256, 0, stream>>>(
        agg, idx, cfg, conf, out, ntiles << 4, nrows);
  }
}